// Model_80350248173925
// MI455X (gfx1250) — compile-verified
//
#include <hip/hip_runtime.h>

typedef float v2f __attribute__((ext_vector_type(2)));
typedef float v8f __attribute__((ext_vector_type(8)));

#define BATCH  64
#define NNODES 1024
#define NEDGES 8192
#define MROWS  (BATCH * NNODES)   // 65536 GEMM rows

#define KC      100               // K-chunk staged in LDS (divides 400/300/100)
#define ASTRIDE 108               // LDS row stride (dwords): ml*44 mod 64 distinct
                                  // multiples of 4 -> conflict-free ds_load_b64

// ===========================================================================
// LDS-tiled GEMM for the big-K layers (K in {400,300,100}; ReLU on input).
// Block = 8 waves sharing a 64-row A panel staged through LDS; wave w owns
// N-tiles (w + 8j), j < NT, inside a (8*NT*16)-column window (blockIdx.y).
// Each wave: 4 M-subtiles x NT N-tiles of V_WMMA_F32_16X16X4_F32.
// ===========================================================================
template <int K, int NF, int NT>
__global__ __launch_bounds__(256) void gemm_lds_wmma(
    const float* __restrict__ X, const float* __restrict__ W,
    const float* __restrict__ bias, float* __restrict__ Y)
{
    __shared__ float shA[64 * ASTRIDE];

    const int tid  = threadIdx.x;
    const int lane = tid & 31;
    const int wave = tid >> 5;
    const int row0 = blockIdx.x * 64;

    const int h  = lane >> 4;     // 0: lanes 0-15, 1: lanes 16-31
    const int ml = lane & 15;     // M idx (A/C) or N idx (B)

    int   nb[NT], nbc[NT];
    float nmask[NT];
    #pragma unroll
    for (int j = 0; j < NT; ++j) {
        const int tile = blockIdx.y * (8 * NT) + wave + 8 * j;
        nb[j]    = tile * 16 + ml;
        nmask[j] = (nb[j] < NF) ? 1.0f : 0.0f;
        nbc[j]   = (nb[j] < NF) ? nb[j] : (NF - 1);    // clamped, in-bounds
    }

    v8f acc[4][NT] = {};

    constexpr int CHUNKS = K / KC;
    for (int ch = 0; ch < CHUNKS; ++ch) {
        const int k0 = ch * KC;

        // ---- cooperative stage of A chunk (64 x KC) with fused ReLU ----
        for (int idx2 = tid; idx2 < 64 * (KC / 2); idx2 += 256) {
            const int r  = idx2 / (KC / 2);
            const int k2 = idx2 - r * (KC / 2);
            v2f v = *(const v2f*)(X + (size_t)(row0 + r) * K + k0 + 2 * k2);
            v.x = fmaxf(v.x, 0.0f);
            v.y = fmaxf(v.y, 0.0f);
            *(v2f*)(&shA[r * ASTRIDE + 2 * k2]) = v;
        }
        __syncthreads();

        // ---- compute on the chunk: KC/4 WMMA k-steps ----
        #pragma unroll 5
        for (int s = 0; s < KC / 4; ++s) {
            const int ka = 4 * s + 2 * h;

            v2f a[4];
            #pragma unroll
            for (int t = 0; t < 4; ++t)
                a[t] = *(const v2f*)(&shA[(16 * t + ml) * ASTRIDE + ka]);

            v2f b[NT];
            #pragma unroll
            for (int j = 0; j < NT; ++j) {
                const int kr = k0 + ka;
                b[j].x = W[(size_t)kr * NF + nbc[j]] * nmask[j];
                b[j].y = W[(size_t)(kr + 1) * NF + nbc[j]] * nmask[j];
            }

            #pragma unroll
            for (int j = 0; j < NT; ++j)
                #pragma unroll
                for (int t = 0; t < 4; ++t)
                    acc[t][j] = __builtin_amdgcn_wmma_f32_16x16x4_f32(
                        false, a[t], false, b[j], (short)0, acc[t][j], false, false);
        }
        __syncthreads();    // protect shA before next stage
    }

    #pragma unroll
    for (int j = 0; j < NT; ++j) {
        if (nb[j] < NF) {
            const float bv = bias[nb[j]];
            #pragma unroll
            for (int t = 0; t < 4; ++t)
                #pragma unroll
                for (int r = 0; r < 8; ++r) {
                    const int mr = row0 + 16 * t + r + 8 * h;
                    Y[(size_t)mr * NF + nb[j]] = acc[t][j][r] + bv;
                }
        }
    }
}

// ===========================================================================
// Simple per-wave GEMM (64x16 tile / wave) for the tiny layers:
//   layer 0: K=2  (A is just H, 512 KB)    layer 5: NF=2 (single N-tile)
// ===========================================================================
template <int K, int NF, bool RELU>
__global__ __launch_bounds__(256) void gemm_bias_wmma(
    const float* __restrict__ X, const float* __restrict__ W,
    const float* __restrict__ bias, float* __restrict__ Y)
{
    constexpr int TILES_N = (NF + 15) / 16;
    constexpr int SUPER_M = MROWS / 64;

    const int lane = threadIdx.x & 31;
    const int wave = threadIdx.x >> 5;
    const int gw   = blockIdx.x * 8 + wave;
    if (gw >= SUPER_M * TILES_N) return;          // wave-uniform

    const int tm   = gw / TILES_N;
    const int tn   = gw - tm * TILES_N;
    const int row0 = tm * 64;
    const int col0 = tn * 16;

    const int h  = lane >> 4;
    const int ml = lane & 15;
    const int nb = col0 + ml;

    constexpr bool FULL_N = (NF % 16) == 0;
    const float nmask = (FULL_N || nb < NF) ? 1.0f : 0.0f;
    const int   nbc   = FULL_N ? nb : (nb < NF ? nb : NF - 1);

    const float* xrow[4];
    #pragma unroll
    for (int t = 0; t < 4; ++t)
        xrow[t] = X + (size_t)(row0 + t * 16 + ml) * K;

    v8f acc[4] = {};

    #pragma unroll 2
    for (int k0 = 0; k0 < K; k0 += 4) {
        const int ka = k0 + 2 * h;

        v2f a[4];
        if constexpr ((K % 4) == 0) {
            #pragma unroll
            for (int t = 0; t < 4; ++t)
                a[t] = *(const v2f*)(xrow[t] + ka);
        } else {
            const int kc = (ka + 2 <= K) ? ka : (K >= 2 ? K - 2 : 0);
            #pragma unroll
            for (int t = 0; t < 4; ++t) {
                v2f tmp = *(const v2f*)(xrow[t] + kc);
                a[t].x = (ka     < K) ? tmp.x : 0.0f;
                a[t].y = (ka + 1 < K) ? tmp.y : 0.0f;
            }
        }
        if constexpr (RELU) {
            #pragma unroll
            for (int t = 0; t < 4; ++t) {
                a[t].x = fmaxf(a[t].x, 0.0f);
                a[t].y = fmaxf(a[t].y, 0.0f);
            }
        }

        v2f b;
        if constexpr ((K % 4) == 0) {
            b.x = W[(size_t)ka * NF + nbc] * nmask;
            b.y = W[(size_t)(ka + 1) * NF + nbc] * nmask;
        } else {
            const int k0c = (ka     < K) ? ka     : K - 1;
            const int k1c = (ka + 1 < K) ? ka + 1 : K - 1;
            b.x = W[(size_t)k0c * NF + nbc] * (nmask * ((ka     < K) ? 1.0f : 0.0f));
            b.y = W[(size_t)k1c * NF + nbc] * (nmask * ((ka + 1 < K) ? 1.0f : 0.0f));
        }

        #pragma unroll
        for (int t = 0; t < 4; ++t)
            acc[t] = __builtin_amdgcn_wmma_f32_16x16x4_f32(
                false, a[t], false, b, (short)0, acc[t], false, false);
    }

    if (FULL_N || nb < NF) {
        const float bv = bias[nb];
        #pragma unroll
        for (int t = 0; t < 4; ++t)
            #pragma unroll
            for (int r = 0; r < 8; ++r) {
                const int mr = row0 + t * 16 + r + 8 * h;
                Y[(size_t)mr * NF + nb] = acc[t][r] + bv;
            }
    }
}

// ---------------------------------------------------------------------------
// agg[b, rows[e], :] += vals[e] * y[b, cols[e], :]   (agg pre-zeroed)
// ---------------------------------------------------------------------------
__global__ void spmm_atomic(const int* __restrict__ rows,
                            const int* __restrict__ cols,
                            const float* __restrict__ vals,
                            const float* __restrict__ Yin,
                            float* __restrict__ Aout, int F)
{
    const int e = blockIdx.x;
    const int b = blockIdx.y;
    const int r = rows[e];
    const int c = cols[e];
    const float v = vals[e];

    const float* src = Yin  + ((size_t)b * NNODES + c) * F;
    float*       dst = Aout + ((size_t)b * NNODES + r) * F;

    for (int f = threadIdx.x; f < F; f += blockDim.x)
        atomicAdd(dst + f, v * src[f]);
}

__global__ void relu_copy(const float* __restrict__ in,
                          float* __restrict__ out, int n)
{
    int i = blockIdx.x * blockDim.x + threadIdx.x;
    if (i < n) out[i] = fmaxf(in[i], 0.0f);
}

// ---------------------------------------------------------------------------
extern "C" void kernel_launch(void* const* d_in, const int* in_sizes, int n_in,
                              void* d_out, int out_size, void* d_ws, size_t ws_size,
                              hipStream_t stream)
{
    (void)in_sizes; (void)n_in; (void)out_size; (void)ws_size;

    static const int DIMS[7] = {2, 400, 300, 100, 300, 400, 2};

    const float* H       = (const float*)d_in[0];
    const int*   rows    = (const int*)  d_in[1];
    const int*   cols    = (const int*)  d_in[2];
    const float* vals_sm = (const float*)d_in[3];
    const float* vals_sp = (const float*)d_in[4];

    const float* Wp[6];
    const float* bp[6];
    for (int i = 0; i < 6; ++i) {
        Wp[i] = (const float*)d_in[5 + 2 * i];
        bp[i] = (const float*)d_in[6 + 2 * i];
    }

    const size_t bufElems = (size_t)MROWS * 400;
    float* bufY = (float*)d_ws;        // GEMM out / SpMM in
    float* bufA = bufY + bufElems;     // SpMM out / next GEMM in

    const int MB = MROWS / 64;         // 1024 M-blocks

    const float* X = H;
    for (int i = 0; i < 6; ++i) {
        const int F = DIMS[i + 1];

        // GEMM: bufY = act(X) @ W + b   (shape-specialized)
        switch (i) {
        case 0:   // K=2: trivial A, simple kernel, no input ReLU
            gemm_bias_wmma<2, 400, false>
                <<<(MB * 25 + 7) / 8, 256, 0, stream>>>(X, Wp[0], bp[0], bufY);
            break;
        case 1:   // tilesN=19 -> NT=3, 1 N-block
            gemm_lds_wmma<400, 300, 3>
                <<<dim3(MB, 1), 256, 0, stream>>>(X, Wp[1], bp[1], bufY);
            break;
        case 2:   // tilesN=7 -> NT=1, 1 N-block
            gemm_lds_wmma<300, 100, 1>
                <<<dim3(MB, 1), 256, 0, stream>>>(X, Wp[2], bp[2], bufY);
            break;
        case 3:   // tilesN=19 -> NT=3, 1 N-block
            gemm_lds_wmma<100, 300, 3>
                <<<dim3(MB, 1), 256, 0, stream>>>(X, Wp[3], bp[3], bufY);
            break;
        case 4:   // tilesN=25 -> NT=2, 2 N-blocks (windows of 16 tiles)
            gemm_lds_wmma<300, 400, 2>
                <<<dim3(MB, 2), 256, 0, stream>>>(X, Wp[4], bp[4], bufY);
            break;
        case 5:   // NF=2: single N-tile, zero A re-reads already
            gemm_bias_wmma<400, 2, true>
                <<<(MB * 1 + 7) / 8, 256, 0, stream>>>(X, Wp[5], bp[5], bufY);
            break;
        }

        // SpMM: bufA = A_sparse * bufY   (zero, then atomic scatter-add)
        hipMemsetAsync(bufA, 0, (size_t)MROWS * F * sizeof(float), stream);
        const float* vals = (i < 3) ? vals_sm : vals_sp;
        dim3 grid(NEDGES, BATCH);
        int th = (F >= 256) ? 256 : (((F + 31) / 32) * 32);
        spmm_atomic<<<grid, th, 0, stream>>>(rows, cols, vals, bufY, bufA, F);

        X = bufA;   // ReLU applied when next layer stages/loads A
    }

    // Final ReLU of the last aggregation (F = 2) into d_out
    const int nOut = MROWS * DIMS[6];
    relu_copy<<<(nOut + 255) / 256, 256, 0, stream>>>(bufA, (float*)d_out, nOut);
}